// LIF_6932077216026
// MI455X (gfx1250) — compile-verified
//
#include <hip/hip_runtime.h>
#include <cstdint>

// LIF scan: B=65536 rows, L=400 timesteps. HBM-bound (~315 MB total traffic
// -> ~13.5us at 23.3 TB/s). Sequential nonlinear recurrence per row => no
// matmul structure (WMMA N/A). CDNA5 paths used: Tensor Data Mover
// (tensor_load_to_lds + s_wait_tensorcnt) staging strided [128 x 40] input
// tiles into LDS with hardware row padding (conflict-free banks), double
// buffered; async-to-LDS inline-asm fallback if the TDM builtin is absent.

constexpr int B_DIM = 65536;
constexpr int L_DIM = 400;
constexpr int ROWS  = 128;            // rows per block == blockDim.x
constexpr int TC    = 40;             // timesteps per chunk
constexpr int NC    = L_DIM / TC;     // 10 chunks

#if defined(__has_builtin)
#if __has_builtin(__builtin_amdgcn_tensor_load_to_lds)
#define HAVE_TDM 1
#endif
#endif

#ifdef HAVE_TDM
// TDM pads 1 DWORD after every 8 DWORDs -> row stride 45 floats (gcd(45,64)=1
// => conflict-free per-step b32 LDS reads across the 32 lanes of a wave).
constexpr int IN_STRIDE = 45;
#else
constexpr int IN_STRIDE = 44;         // 176B, 16B-aligned for b128 path
constexpr int LOADS_PER_THREAD = (ROWS * TC / 4) / ROWS; // 10 b128 loads/thread/chunk
#endif

// ---- wait primitives -------------------------------------------------------
#if defined(__has_builtin)
#if __has_builtin(__builtin_amdgcn_s_wait_tensorcnt)
#define WAIT_TENSOR(N) __builtin_amdgcn_s_wait_tensorcnt(N)
#endif
#if __has_builtin(__builtin_amdgcn_s_wait_asynccnt)
#define WAIT_ASYNC(N) __builtin_amdgcn_s_wait_asynccnt(N)
#endif
#endif
#ifndef WAIT_TENSOR
#define WAIT_TENSOR(N) asm volatile("s_wait_tensorcnt %0" :: "i"(N) : "memory")
#endif
#ifndef WAIT_ASYNC
#define WAIT_ASYNC(N) asm volatile("s_wait_asynccnt %0" :: "i"(N) : "memory")
#endif

// 16B async copy: global (64-bit VGPR pair) -> LDS (32-bit byte address VGPR).
#define ASYNC_LOAD_B128(LDS_U32, GPTR)                              \
  asm volatile("global_load_async_to_lds_b128 %0, %1, off"          \
               :: "v"(LDS_U32), "v"(GPTR) : "memory")

#ifdef HAVE_TDM
typedef unsigned int u32x4 __attribute__((ext_vector_type(4)));
typedef int          i32x8 __attribute__((ext_vector_type(8)));
typedef int          i32x4 __attribute__((ext_vector_type(4)));

// One TDM descriptor moves the whole [ROWS x TC] f32 tile (row stride L_DIM
// elements in global) into LDS with 1-DWORD padding every 8 DWORDs.
__device__ __forceinline__ void tdm_prefetch(const float* gtile, unsigned lds_addr) {
  const uint64_t ga = (uint64_t)(uintptr_t)gtile;
  // D# group 0: count=1 (valid, user), lds_addr, global_addr[56:0], type=2.
  u32x4 g0 = { 1u,
               lds_addr,
               (unsigned)ga,
               (unsigned)(ga >> 32) | 0x80000000u };
  // D# group 1:
  //  w0: data_size=2 (4B)<<16 | pad_enable<<20 | pad_interval=2 (8 DW)<<22
  //      | pad_amount=0 (1 DW)<<25 ; workgroup_mask=0 (not in cluster)
  //  w1: tensor_dim0[15:0]=400 at bits[63:48]
  //  w3: tensor_dim1[31:16]=1 (65536 rows) | tile_dim0=40 at bits[127:112]
  //  w4: tile_dim1=128 rows ; tile_dim2=0 (unused -> 2D tile)
  //  w5: tensor_dim0_stride=400 elements
  i32x8 g1 = { (int)0x00920000,
               (int)0x01900000,
               0,
               (int)0x00280001,
               (int)ROWS,
               (int)L_DIM,
               0, 0 };
  i32x4 gz4 = { 0, 0, 0, 0 };
  i32x8 gz8 = { 0, 0, 0, 0, 0, 0, 0, 0 };
  // amdgpu-toolchain (clang-23) 6-arg form:
  // (g0[4 SGPR], g1[8 SGPR], g2[4 SGPR], g3[4 SGPR], extra[8], cpol)
  __builtin_amdgcn_tensor_load_to_lds(g0, g1, gz4, gz4, gz8, 0);
}
#endif

__global__ __launch_bounds__(ROWS)
void lif_scan_kernel(const float* __restrict__ I,
                     float* __restrict__ spikes,
                     float* __restrict__ series,
                     float* __restrict__ first_out)
{
  __shared__ __align__(16) float lin[2][ROWS * IN_STRIDE];

  const int tid     = threadIdx.x;
  const int rowBase = blockIdx.x * ROWS;
  const int myRow   = rowBase + tid;

  const char* gbase = (const char*)(I + (size_t)rowBase * L_DIM);
  const unsigned lbase[2] = { (unsigned)(uintptr_t)&lin[0][0],
                              (unsigned)(uintptr_t)&lin[1][0] };

#ifdef HAVE_TDM
  const bool wave0 = (tid < 32);
  auto prefetch = [&](int c, unsigned lb) {
    tdm_prefetch((const float*)(gbase + (unsigned)(c * TC * 4)), lb);
  };
#else
  // lane -> (row, quad) mapping for the chunk tile [ROWS x TC] floats.
  unsigned goff[LOADS_PER_THREAD];
  unsigned loff[LOADS_PER_THREAD];
#pragma unroll
  for (int i = 0; i < LOADS_PER_THREAD; ++i) {
    unsigned p    = (unsigned)(i * ROWS + tid);
    unsigned row  = p / 10u;
    unsigned quad = p - row * 10u;
    goff[i] = row * (unsigned)(L_DIM * 4) + quad * 16u;
    loff[i] = row * (unsigned)(IN_STRIDE * 4) + quad * 16u;
  }
  auto prefetch = [&](int c, unsigned lb) {
    const char* cb = gbase + (unsigned)(c * TC * 4);
#pragma unroll
    for (int i = 0; i < LOADS_PER_THREAD; ++i) {
      const void* gp = cb + goff[i];
      unsigned la = lb + loff[i];
      ASYNC_LOAD_B128(la, gp);
    }
  };
#endif

  // Recurrence state.
  constexpr float KEEP_V = 1.0f - 1.0f / 20.0f;    // 1 - 1/TAU
  constexpr float KEEP_A = 1.0f - 1.0f / 100.0f;   // 1 - 1/TAU_ADAPT
  float v = 0.0f, a = 0.0f, csum = 0.0f;
  int first = L_DIM;

#ifdef HAVE_TDM
  if (wave0) prefetch(0, lbase[0]);
#else
  prefetch(0, lbase[0]);
#endif

  for (int c = 0; c < NC; ++c) {
#ifdef HAVE_TDM
    if (wave0) {
      if (c + 1 < NC) {
        prefetch(c + 1, lbase[(c + 1) & 1]);
        WAIT_TENSOR(1);               // chunk c's TDM done (in-order per wave)
      } else {
        WAIT_TENSOR(0);
      }
    }
#else
    if (c + 1 < NC) {
      prefetch(c + 1, lbase[(c + 1) & 1]);
      WAIT_ASYNC(LOADS_PER_THREAD);
    } else {
      WAIT_ASYNC(0);
    }
#endif
    __syncthreads();                  // chunk-c tile visible to all waves

    const float* inRow = &lin[c & 1][tid * IN_STRIDE];
    float* spRow = spikes + (size_t)myRow * L_DIM + c * TC;
    float* seRow = series + (size_t)myRow * L_DIM + c * TC;

#pragma unroll
    for (int tq = 0; tq < TC / 4; ++tq) {
      float xs[4];
#pragma unroll
      for (int k = 0; k < 4; ++k) {
        int tl = tq * 4 + k;
#ifdef HAVE_TDM
        xs[k] = inRow[(tl >> 3) * 9 + (tl & 7)];   // padded 8+1 layout
#else
        xs[k] = inRow[tl];
#endif
      }
      float ss[4], cs[4];
#pragma unroll
      for (int k = 0; k < 4; ++k) {
        v = fmaf(v, KEEP_V, xs[k]);            // v += -v/tau + I_t
        float th = fmaf(1.5f, a, 1.0f);        // TH + BETA_ADAPT*a
        bool sp = (v >= th);
        float s = sp ? 1.0f : 0.0f;            // forward value of ST spike
        v = sp ? -0.5f : v;                    // hard reset
        a = fmaf(a, KEEP_A, s);                // a += -a/tau_adapt + s
        csum += s;
        int t = c * TC + tq * 4 + k;
        if (sp && first == L_DIM) first = t;   // first hard spike index
        ss[k] = s;
        cs[k] = csum;
      }
      *(float4*)&spRow[tq * 4] = make_float4(ss[0], ss[1], ss[2], ss[3]);
      *(float4*)&seRow[tq * 4] = make_float4(cs[0], cs[1], cs[2], cs[3]);
    }
    __syncthreads();                  // buffer (c&1) reusable at iteration c+2
  }

  // hard_first is int32 in the reference; values <= 400 are exact in f32.
  first_out[myRow] = (float)first;
}

extern "C" void kernel_launch(void* const* d_in, const int* in_sizes, int n_in,
                              void* d_out, int out_size, void* d_ws, size_t ws_size,
                              hipStream_t stream) {
  const float* I = (const float*)d_in[0];
  float* out    = (float*)d_out;
  float* spikes = out;                                   // [B, L]
  float* series = out + (size_t)B_DIM * L_DIM;           // [B, L]
  float* first  = out + (size_t)2 * B_DIM * L_DIM;       // [B]
  lif_scan_kernel<<<B_DIM / ROWS, ROWS, 0, stream>>>(I, spikes, series, first);
}